// SMPL_55843164782846
// MI455X (gfx1250) — compile-verified
//
#include <hip/hip_runtime.h>

#define B_  1024
#define V_  6890
#define J_  24
#define NB_ 10
#define KP_ 207          // pose basis = 9*(J-1)
#define N_  (V_ * 3)     // 20670 columns of the big GEMM

typedef float v2f __attribute__((ext_vector_type(2)));
typedef float v8f __attribute__((ext_vector_type(8)));

static __device__ __forceinline__ v8f wmma_f32(v2f a, v2f b, v8f c) {
  return __builtin_amdgcn_wmma_f32_16x16x4_f32(false, a, false, b, (short)0, c,
                                               false, false);
}

// ---------------------------------------------------------------------------
// Kernel A: collapse J_regressor through the shape space.
//   jr_vt[j,c]   = sum_v JR[j,v] * v_template[v,c]              (24x3)
//   jr_sd[j,c,l] = sum_v JR[j,v] * shapedirs[v,c,l]             (24x3x10)
// ---------------------------------------------------------------------------
__global__ __launch_bounds__(256)
void jreg_kernel(const float* __restrict__ JR, const float* __restrict__ vtempl,
                 const float* __restrict__ sdirs,
                 float* __restrict__ jr_sd, float* __restrict__ jr_vt) {
  const int j = blockIdx.x / 3, c = blockIdx.x % 3;
  float part[NB_ + 1];
#pragma unroll
  for (int i = 0; i <= NB_; ++i) part[i] = 0.f;
  for (int v = threadIdx.x; v < V_; v += 256) {
    const float wjv = JR[j * V_ + v];
    part[NB_] += wjv * vtempl[v * 3 + c];
#pragma unroll
    for (int l = 0; l < NB_; ++l)
      part[l] += wjv * sdirs[(v * 3 + c) * NB_ + l];
  }
  __shared__ float red[256];
  for (int comp = 0; comp <= NB_; ++comp) {
    red[threadIdx.x] = part[comp];
    __syncthreads();
    for (int s = 128; s > 0; s >>= 1) {
      if (threadIdx.x < s) red[threadIdx.x] += red[threadIdx.x + s];
      __syncthreads();
    }
    if (threadIdx.x == 0) {
      if (comp == NB_) jr_vt[j * 3 + c] = red[0];
      else             jr_sd[(j * 3 + c) * NB_ + comp] = red[0];
    }
    __syncthreads();
  }
}

// ---------------------------------------------------------------------------
// Kernel B: per-batch joint positions + kinematic chain + relative transforms.
// One thread per batch. Writes posed_joints (output 2) and A_rel (3x4 rows).
// ---------------------------------------------------------------------------
__global__ __launch_bounds__(64)
void chain_kernel(const float* __restrict__ betas, const float* __restrict__ grot,
                  const float* __restrict__ bpose,
                  const float* __restrict__ jr_sd, const float* __restrict__ jr_vt,
                  float* __restrict__ a_rel, float* __restrict__ out_joints) {
  const int b = blockIdx.x * blockDim.x + threadIdx.x;
  if (b >= B_) return;
  const int par[J_] = {-1,0,0,0,1,2,3,4,5,6,7,8,9,9,9,12,13,14,16,17,18,19,20,21};

  float Jt[J_][3];
#pragma unroll
  for (int j = 0; j < J_; ++j)
#pragma unroll
    for (int c = 0; c < 3; ++c) {
      float acc = jr_vt[j * 3 + c];
#pragma unroll
      for (int l = 0; l < NB_; ++l)
        acc += betas[b * NB_ + l] * jr_sd[(j * 3 + c) * NB_ + l];
      Jt[j][c] = acc;
    }

  float R[J_][9];
  float t[J_][3];
#pragma unroll
  for (int j = 0; j < J_; ++j) {
    float lr[9];
    const float* src = (j == 0) ? (grot + (size_t)b * 9)
                                : (bpose + (size_t)b * KP_ + (j - 1) * 9);
#pragma unroll
    for (int i = 0; i < 9; ++i) lr[i] = src[i];

    if (j == 0) {
#pragma unroll
      for (int i = 0; i < 9; ++i) R[0][i] = lr[i];
#pragma unroll
      for (int c = 0; c < 3; ++c) t[0][c] = Jt[0][c];
    } else {
      const int p = par[j];
      float lt[3];
#pragma unroll
      for (int c = 0; c < 3; ++c) lt[c] = Jt[j][c] - Jt[p][c];
#pragma unroll
      for (int r = 0; r < 3; ++r) {
#pragma unroll
        for (int c = 0; c < 3; ++c)
          R[j][r * 3 + c] = R[p][r * 3 + 0] * lr[0 + c]
                          + R[p][r * 3 + 1] * lr[3 + c]
                          + R[p][r * 3 + 2] * lr[6 + c];
        t[j][r] = R[p][r * 3 + 0] * lt[0] + R[p][r * 3 + 1] * lt[1]
                + R[p][r * 3 + 2] * lt[2] + t[p][r];
      }
    }
#pragma unroll
    for (int c = 0; c < 3; ++c)
      out_joints[((size_t)b * J_ + j) * 3 + c] = t[j][c];

    float* dst = a_rel + ((size_t)b * J_ + j) * 12;
#pragma unroll
    for (int r = 0; r < 3; ++r) {
      const float rt = t[j][r] - (R[j][r * 3 + 0] * Jt[j][0] +
                                  R[j][r * 3 + 1] * Jt[j][1] +
                                  R[j][r * 3 + 2] * Jt[j][2]);
      dst[r * 4 + 0] = R[j][r * 3 + 0];
      dst[r * 4 + 1] = R[j][r * 3 + 1];
      dst[r * 4 + 2] = R[j][r * 3 + 2];
      dst[r * 4 + 3] = rt;
    }
  }
}

// ---------------------------------------------------------------------------
// Kernel C: v_posed = (full_pose[:,1:] - I) @ posedirs + v_template
//                     + shapedirs @ betas   (fused epilogue)
// WMMA f32 16x16x4, branch-free main loop.
// Wave tile = 16 batches x 64 cols; 8 waves/block.
// ---------------------------------------------------------------------------
__global__ __launch_bounds__(256)
void posed_gemm_kernel(const float* __restrict__ bpose, const float* __restrict__ pdirs,
                       const float* __restrict__ betas, const float* __restrict__ vtempl,
                       const float* __restrict__ sdirs, float* __restrict__ v_posed) {
  __shared__ float lds_pf[16][208];   // K padded 207 -> 208, col 207 = 0
  __shared__ float lds_beta[16][NB_];
  const int tid   = threadIdx.x;
  const int mbase = blockIdx.y * 16;

  for (int idx = tid; idx < 16 * 208; idx += 256) {
    const int m = idx / 208, kk = idx % 208;
    float v = 0.f;
    if (kk < KP_) {
      v = bpose[(size_t)(mbase + m) * KP_ + kk];
      if ((kk % 9) % 4 == 0) v -= 1.f;     // subtract identity on the diagonal
    }
    lds_pf[m][kk] = v;
  }
  for (int idx = tid; idx < 16 * NB_; idx += 256)
    lds_beta[idx / NB_][idx % NB_] = betas[(size_t)(mbase + idx / NB_) * NB_ + idx % NB_];
  __syncthreads();

  const int wave = tid >> 5;
  const int lane = tid & 31;
  const int half = lane >> 4;
  const int lm   = lane & 15;
  const int colbase = (blockIdx.x * 8 + wave) * 64;
  if (colbase >= N_) return;             // wave-uniform

  // Loop-invariant clamped columns: OOB lanes compute garbage columns that
  // are never stored (B column n only feeds output column n).
  int col[4], cc[4];
#pragma unroll
  for (int tn = 0; tn < 4; ++tn) {
    col[tn] = colbase + tn * 16 + lm;
    cc[tn]  = col[tn] < N_ ? col[tn] : (N_ - 1);
  }

  const v8f vzero = {0.f,0.f,0.f,0.f,0.f,0.f,0.f,0.f};
  v8f acc0 = vzero, acc1 = vzero, acc2 = vzero, acc3 = vzero;

  // lanes 0-15 read K rows k,k+1 ; lanes 16-31 read rows k+2,k+3
  const float* pk = pdirs + (size_t)(2 * half) * N_;
#pragma unroll 2
  for (int k = 0; k < 204; k += 4) {     // 51 guard-free steps (rows <= 203)
    const int ka = k + 2 * half;
    v2f a; a.x = lds_pf[lm][ka]; a.y = lds_pf[lm][ka + 1];
    v2f b0, b1, b2, b3;
    b0.x = pk[cc[0]]; b0.y = pk[(size_t)N_ + cc[0]];
    b1.x = pk[cc[1]]; b1.y = pk[(size_t)N_ + cc[1]];
    b2.x = pk[cc[2]]; b2.y = pk[(size_t)N_ + cc[2]];
    b3.x = pk[cc[3]]; b3.y = pk[(size_t)N_ + cc[3]];
    acc0 = wmma_f32(a, b0, acc0);
    acc1 = wmma_f32(a, b1, acc1);
    acc2 = wmma_f32(a, b2, acc2);
    acc3 = wmma_f32(a, b3, acc3);
    pk += 4 * (size_t)N_;
  }
  { // k = 204 tail: rows 204..206 valid; A K=207 is the LDS zero pad, so the
    // matching B row index can be clamped to 206 with zero contribution.
    const int ka = 204 + 2 * half;
    v2f a; a.x = lds_pf[lm][ka]; a.y = lds_pf[lm][ka + 1];
    const size_t r0 = (size_t)ka * N_;
    const size_t r1 = (size_t)((ka + 1 < KP_) ? ka + 1 : KP_ - 1) * N_;
    v2f b0, b1, b2, b3;
    b0.x = pdirs[r0 + cc[0]]; b0.y = pdirs[r1 + cc[0]];
    b1.x = pdirs[r0 + cc[1]]; b1.y = pdirs[r1 + cc[1]];
    b2.x = pdirs[r0 + cc[2]]; b2.y = pdirs[r1 + cc[2]];
    b3.x = pdirs[r0 + cc[3]]; b3.y = pdirs[r1 + cc[3]];
    acc0 = wmma_f32(a, b0, acc0);
    acc1 = wmma_f32(a, b1, acc1);
    acc2 = wmma_f32(a, b2, acc2);
    acc3 = wmma_f32(a, b3, acc3);
  }

  // epilogue: add v_template + shapedirs @ betas, store v_posed
  auto epi = [&](int tn, const v8f& a8) {
    if (col[tn] >= N_) return;
    const float vt = vtempl[col[tn]];
    float sd[NB_];
#pragma unroll
    for (int l = 0; l < NB_; ++l) sd[l] = sdirs[(size_t)col[tn] * NB_ + l];
#pragma unroll
    for (int g = 0; g < 8; ++g) {
      const int mloc = g + 8 * half;     // C layout: VGPR g -> row g (+8 upper half)
      float add = vt;
#pragma unroll
      for (int l = 0; l < NB_; ++l) add += lds_beta[mloc][l] * sd[l];
      v_posed[(size_t)(mbase + mloc) * N_ + col[tn]] = a8[g] + add;
    }
  };
  epi(0, acc0); epi(1, acc1); epi(2, acc2); epi(3, acc3);
}

// ---------------------------------------------------------------------------
// Kernel D: LBS blend on WMMA.
//   T[v,0:12] = W[v,0:24] @ A_rel[b]   -> (16x24)@(24x12) per (vtile, batch)
// 8 waves/block: one batch each, same vertex tile. K=24 -> 6 wmma steps.
// A_rel staged transposed in LDS so B fragments are single ds_load_b64.
// T bounced through LDS (wave-local) for the per-vertex affine transform.
// ---------------------------------------------------------------------------
__global__ __launch_bounds__(256)
void blend_wmma_kernel(const float* __restrict__ a_rel, const float* __restrict__ w,
                       const float* __restrict__ v_posed, float* __restrict__ out) {
  __shared__ float lds_At[8][16][24];   // [wave-batch][T component][joint], rows 12-15 = 0
  __shared__ float lds_T[8][16][13];    // bounce: 16 vertices x 12 comps (+pad)
  const int tid = threadIdx.x;
  const int bg  = blockIdx.y * 8;

  for (int idx = tid; idx < 8 * 16 * J_; idx += 256) {
    const int wb = idx / (16 * J_);
    const int rem = idx - wb * 16 * J_;
    const int comp = rem / J_, kk = rem % J_;
    lds_At[wb][comp][kk] =
        (comp < 12) ? a_rel[((size_t)(bg + wb) * J_ + kk) * 12 + comp] : 0.f;
  }
  __syncthreads();

  const int wave = tid >> 5;
  const int lane = tid & 31;
  const int half = lane >> 4;
  const int lm   = lane & 15;
  const int b     = bg + wave;
  const int vbase = blockIdx.x * 16;
  int vr = vbase + lm; if (vr >= V_) vr = V_ - 1;   // clamp ragged last tile

  const v8f vzero = {0.f,0.f,0.f,0.f,0.f,0.f,0.f,0.f};
  v8f acc = vzero;
#pragma unroll
  for (int ks = 0; ks < 6; ++ks) {
    const int ka = 4 * ks + 2 * half;
    v2f a;                                // A = lbs_weights rows (vertices)
    a.x = w[(size_t)vr * J_ + ka];
    a.y = w[(size_t)vr * J_ + ka + 1];
    v2f bf;                               // B = A_rel[b] (24 x 12, transposed in LDS)
    bf.x = lds_At[wave][lm][ka];
    bf.y = lds_At[wave][lm][ka + 1];
    acc = wmma_f32(a, bf, acc);
  }

  // bounce T tile through LDS (wave-local; same-wave DS ops stay in order)
  if (lm < 12) {
#pragma unroll
    for (int g = 0; g < 8; ++g)
      lds_T[wave][g + 8 * half][lm] = acc[g];
  }

  if (half == 0) {                        // lanes 0-15: one vertex each
    const int v = vbase + lm;
    if (v < V_) {
      float T[12];
#pragma unroll
      for (int i = 0; i < 12; ++i) T[i] = lds_T[wave][lm][i];
      const size_t base = (size_t)b * N_ + (size_t)v * 3;
      const float x = v_posed[base + 0];
      const float y = v_posed[base + 1];
      const float z = v_posed[base + 2];
      out[base + 0] = T[0] * x + T[1] * y + T[2]  * z + T[3];
      out[base + 1] = T[4] * x + T[5] * y + T[6]  * z + T[7];
      out[base + 2] = T[8] * x + T[9] * y + T[10] * z + T[11];
    }
  }
}

// ---------------------------------------------------------------------------
extern "C" void kernel_launch(void* const* d_in, const int* in_sizes, int n_in,
                              void* d_out, int out_size, void* d_ws, size_t ws_size,
                              hipStream_t stream) {
  const float* betas  = (const float*)d_in[0];
  const float* grot   = (const float*)d_in[1];
  const float* bpose  = (const float*)d_in[2];
  const float* vtempl = (const float*)d_in[3];
  const float* sdirs  = (const float*)d_in[4];
  const float* pdirs  = (const float*)d_in[5];
  const float* jreg   = (const float*)d_in[6];
  const float* lbsw   = (const float*)d_in[7];

  float* out_vert   = (float*)d_out;
  float* out_joints = out_vert + (size_t)B_ * V_ * 3;

  float* ws     = (float*)d_ws;
  float* jr_sd  = ws;                               // 720 floats
  float* jr_vt  = ws + 720;                         // 72 floats
  float* a_rel  = ws + 1024;                        // B*24*12 = 294912 floats
  float* vposed = ws + 1024 + (size_t)B_ * J_ * 12; // B*V*3   = 21.2M floats

  jreg_kernel<<<72, 256, 0, stream>>>(jreg, vtempl, sdirs, jr_sd, jr_vt);
  chain_kernel<<<B_ / 64, 64, 0, stream>>>(betas, grot, bpose, jr_sd, jr_vt,
                                           a_rel, out_joints);
  posed_gemm_kernel<<<dim3(41, 64), 256, 0, stream>>>(bpose, pdirs, betas,
                                                      vtempl, sdirs, vposed);
  blend_wmma_kernel<<<dim3((V_ + 15) / 16, B_ / 8), 256, 0, stream>>>(
      a_rel, lbsw, vposed, out_vert);
}